// DMPNNEncoder_86672440033885
// MI455X (gfx1250) — compile-verified
//
#include <hip/hip_runtime.h>
#include <hip/hip_bf16.h>
#include <math.h>

typedef __attribute__((ext_vector_type(2))) float v2f;
typedef __attribute__((ext_vector_type(8))) float v8f;

#define F_DIM 133   // node feature dim
#define HID   128   // hidden
#define EMB_D 64    // embedding out

__device__ __forceinline__ v8f wmma_f32_k4(v2f a, v2f b, v8f c) {
  // D = A(16x4, f32) * B(4x16, f32) + C(16x16, f32)
  return __builtin_amdgcn_wmma_f32_16x16x4_f32(
      /*neg_a=*/false, a, /*neg_b=*/false, b,
      /*c_mod=*/(short)0, c, /*reuse_a=*/false, /*reuse_b=*/false);
}

// ---------------------------------------------------------------- utilities
__global__ void zero_f32(float* __restrict__ p, int n) {
  int i = blockIdx.x * blockDim.x + threadIdx.x;
  if (i < n) p[i] = 0.f;
}

// Transpose + zero-pad weights into [K][N] layout for coalesced B loads.
__global__ void prep_weights(const float* __restrict__ W1, const float* __restrict__ W2,
                             const float* __restrict__ W3, const float* __restrict__ Wfc,
                             float* __restrict__ W1t, float* __restrict__ W2t,
                             float* __restrict__ W3t, float* __restrict__ Wfct) {
  int i = blockIdx.x * blockDim.x + threadIdx.x;
  if (i < 136 * HID) {                 // W1t[136][128], K padded 134->136
    int k = i >> 7, j = i & 127;
    W1t[i] = (k < F_DIM + 1) ? W1[j * (F_DIM + 1) + k] : 0.f;
  }
  if (i < HID * HID) {                 // W2t[128][128]
    int k = i >> 7, j = i & 127;
    W2t[i] = W2[j * HID + k];
  }
  if (i < 264 * HID) {                 // W3t[264][128], K padded 261->264
    int k = i >> 7, j = i & 127;
    W3t[i] = (k < F_DIM + HID) ? W3[j * (F_DIM + HID) + k] : 0.f;
  }
  if (i < HID * EMB_D) {               // Wfct[128][64]
    int k = i / EMB_D, j = i % EMB_D;
    Wfct[i] = Wfc[j * HID + k];
  }
}

// ------------------------------------------------- h0 = relu([x[src],ea] @ W1t)
// block: 256 threads (8 waves), 64 edges per block (4 M-subtiles),
// wave w -> cols [16w,16w+16); one B fragment feeds 4 WMMAs.
__global__ void edge_init_gemm(const float* __restrict__ x, const int* __restrict__ src,
                               const float* __restrict__ eattr, const float* __restrict__ W1t,
                               float* __restrict__ h0, int E) {
  __shared__ float As[64 * 140];  // 64 edges x 136 (padded stride 140)
  const int tid = threadIdx.x;
  const int e0 = blockIdx.x * 64;
  for (int i = tid; i < 64 * 136; i += 256) {
    int r = i / 136, k = i - r * 136;
    int e = e0 + r;
    float v = 0.f;
    if (e < E) {
      if (k < F_DIM) v = x[(size_t)src[e] * F_DIM + k];
      else if (k == F_DIM) v = eattr[e];
    }
    As[r * 140 + k] = v;
  }
  __syncthreads();
  const int lane = tid & 31, wave = tid >> 5;
  const int half = lane >> 4, lm = lane & 15;
  const int ncol = wave * 16 + lm;
  v8f c0 = {}, c1 = {}, c2 = {}, c3 = {};
#pragma unroll
  for (int kk = 0; kk < 136; kk += 4) {
    int kr = kk + 2 * half;
    v2f b;
    b.x = W1t[kr * HID + ncol];
    b.y = W1t[(kr + 1) * HID + ncol];
    v2f a0, a1, a2, a3;
    a0.x = As[(lm +  0) * 140 + kr]; a0.y = As[(lm +  0) * 140 + kr + 1];
    a1.x = As[(lm + 16) * 140 + kr]; a1.y = As[(lm + 16) * 140 + kr + 1];
    a2.x = As[(lm + 32) * 140 + kr]; a2.y = As[(lm + 32) * 140 + kr + 1];
    a3.x = As[(lm + 48) * 140 + kr]; a3.y = As[(lm + 48) * 140 + kr + 1];
    c0 = wmma_f32_k4(a0, b, c0);
    c1 = wmma_f32_k4(a1, b, c1);
    c2 = wmma_f32_k4(a2, b, c2);
    c3 = wmma_f32_k4(a3, b, c3);
  }
#pragma unroll
  for (int r = 0; r < 8; ++r) {
    int m = r + 8 * half;
    int e;
    e = e0 + m;      if (e < E) h0[(size_t)e * HID + ncol] = fmaxf(c0[r], 0.f);
    e = e0 + m + 16; if (e < E) h0[(size_t)e * HID + ncol] = fmaxf(c1[r], 0.f);
    e = e0 + m + 32; if (e < E) h0[(size_t)e * HID + ncol] = fmaxf(c2[r], 0.f);
    e = e0 + m + 48; if (e < E) h0[(size_t)e * HID + ncol] = fmaxf(c3[r], 0.f);
  }
}

// ----------------------------------------------- acc[dst[e]] += h[e]  (atomic)
__global__ void edge_segsum(const float* __restrict__ h, const int* __restrict__ dst,
                            float* __restrict__ acc, int E) {
  int t = blockIdx.x * blockDim.x + threadIdx.x;
  int e = t >> 5;
  if (e >= E) return;
  int c4 = (t & 31) << 2;
  const float4 v = *(const float4*)(h + (size_t)e * HID + c4);
  float* p = acc + (size_t)dst[e] * HID + c4;
  atomicAdd(p + 0, v.x);
  atomicAdd(p + 1, v.y);
  atomicAdd(p + 2, v.z);
  atomicAdd(p + 3, v.w);
}

// ------- h_out = relu(h0 + (m_node[src]-h_in[rev]) @ W2t), 64 edges per block
__global__ void edge_msg_gemm(const float* __restrict__ h_in, const float* __restrict__ m_node,
                              const int* __restrict__ src, const int* __restrict__ rev,
                              const float* __restrict__ W2t, const float* __restrict__ h0,
                              float* __restrict__ h_out, int E) {
  __shared__ float As[64 * 132];  // 64 edges x 128 (padded stride 132)
  const int tid = threadIdx.x;
  const int e0 = blockIdx.x * 64;
  // vectorized staging: thread handles 4 consecutive k
  for (int i = tid; i < 64 * 32; i += 256) {
    int r = i >> 5, k4 = (i & 31) << 2;
    int e = e0 + r;
    float4 v = make_float4(0.f, 0.f, 0.f, 0.f);
    if (e < E) {
      const float4 mv = *(const float4*)(m_node + (size_t)src[e] * HID + k4);
      const float4 hv = *(const float4*)(h_in + (size_t)rev[e] * HID + k4);
      v = make_float4(mv.x - hv.x, mv.y - hv.y, mv.z - hv.z, mv.w - hv.w);
    }
    *(float4*)(&As[r * 132 + k4]) = v;
  }
  __syncthreads();
  const int lane = tid & 31, wave = tid >> 5;
  const int half = lane >> 4, lm = lane & 15;
  const int ncol = wave * 16 + lm;
  v8f c0 = {}, c1 = {}, c2 = {}, c3 = {};
#pragma unroll
  for (int kk = 0; kk < HID; kk += 4) {
    int kr = kk + 2 * half;
    v2f b;
    b.x = W2t[kr * HID + ncol];
    b.y = W2t[(kr + 1) * HID + ncol];
    v2f a0, a1, a2, a3;
    a0.x = As[(lm +  0) * 132 + kr]; a0.y = As[(lm +  0) * 132 + kr + 1];
    a1.x = As[(lm + 16) * 132 + kr]; a1.y = As[(lm + 16) * 132 + kr + 1];
    a2.x = As[(lm + 32) * 132 + kr]; a2.y = As[(lm + 32) * 132 + kr + 1];
    a3.x = As[(lm + 48) * 132 + kr]; a3.y = As[(lm + 48) * 132 + kr + 1];
    c0 = wmma_f32_k4(a0, b, c0);
    c1 = wmma_f32_k4(a1, b, c1);
    c2 = wmma_f32_k4(a2, b, c2);
    c3 = wmma_f32_k4(a3, b, c3);
  }
#pragma unroll
  for (int r = 0; r < 8; ++r) {
    int m = r + 8 * half;
    int e;
    e = e0 + m;
    if (e < E) h_out[(size_t)e * HID + ncol] = fmaxf(h0[(size_t)e * HID + ncol] + c0[r], 0.f);
    e = e0 + m + 16;
    if (e < E) h_out[(size_t)e * HID + ncol] = fmaxf(h0[(size_t)e * HID + ncol] + c1[r], 0.f);
    e = e0 + m + 32;
    if (e < E) h_out[(size_t)e * HID + ncol] = fmaxf(h0[(size_t)e * HID + ncol] + c2[r], 0.f);
    e = e0 + m + 48;
    if (e < E) h_out[(size_t)e * HID + ncol] = fmaxf(h0[(size_t)e * HID + ncol] + c3[r], 0.f);
  }
}

// ---- node_attr = relu([x, v_msg] @ W3t + b3); gsums[batch[n]] += node_attr[n]
// 64 nodes per block
__global__ void node_gemm_pool(const float* __restrict__ x, const float* __restrict__ vmsg,
                               const float* __restrict__ W3t, const float* __restrict__ b3,
                               const int* __restrict__ batch, float* __restrict__ gsums, int N) {
  __shared__ float As[64 * 268];  // 64 nodes x 264 (padded stride 268)
  const int tid = threadIdx.x;
  const int n0 = blockIdx.x * 64;
  for (int i = tid; i < 64 * 264; i += 256) {
    int r = i / 264, k = i - r * 264;
    int n = n0 + r;
    float v = 0.f;
    if (n < N) {
      if (k < F_DIM) v = x[(size_t)n * F_DIM + k];
      else if (k < F_DIM + HID) v = vmsg[(size_t)n * HID + (k - F_DIM)];
    }
    As[r * 268 + k] = v;
  }
  __syncthreads();
  const int lane = tid & 31, wave = tid >> 5;
  const int half = lane >> 4, lm = lane & 15;
  const int ncol = wave * 16 + lm;
  v8f c0 = {}, c1 = {}, c2 = {}, c3 = {};
#pragma unroll
  for (int kk = 0; kk < 264; kk += 4) {
    int kr = kk + 2 * half;
    v2f b;
    b.x = W3t[kr * HID + ncol];
    b.y = W3t[(kr + 1) * HID + ncol];
    v2f a0, a1, a2, a3;
    a0.x = As[(lm +  0) * 268 + kr]; a0.y = As[(lm +  0) * 268 + kr + 1];
    a1.x = As[(lm + 16) * 268 + kr]; a1.y = As[(lm + 16) * 268 + kr + 1];
    a2.x = As[(lm + 32) * 268 + kr]; a2.y = As[(lm + 32) * 268 + kr + 1];
    a3.x = As[(lm + 48) * 268 + kr]; a3.y = As[(lm + 48) * 268 + kr + 1];
    c0 = wmma_f32_k4(a0, b, c0);
    c1 = wmma_f32_k4(a1, b, c1);
    c2 = wmma_f32_k4(a2, b, c2);
    c3 = wmma_f32_k4(a3, b, c3);
  }
  const float bias = b3[ncol];
#pragma unroll
  for (int r = 0; r < 8; ++r) {
    int m = r + 8 * half;
    int n;
    n = n0 + m;
    if (n < N) atomicAdd(&gsums[(size_t)batch[n] * HID + ncol], fmaxf(c0[r] + bias, 0.f));
    n = n0 + m + 16;
    if (n < N) atomicAdd(&gsums[(size_t)batch[n] * HID + ncol], fmaxf(c1[r] + bias, 0.f));
    n = n0 + m + 32;
    if (n < N) atomicAdd(&gsums[(size_t)batch[n] * HID + ncol], fmaxf(c2[r] + bias, 0.f));
    n = n0 + m + 48;
    if (n < N) atomicAdd(&gsums[(size_t)batch[n] * HID + ncol], fmaxf(c3[r] + bias, 0.f));
  }
}

__global__ void node_counts(const int* __restrict__ batch, float* __restrict__ counts, int N) {
  int n = blockIdx.x * blockDim.x + threadIdx.x;
  if (n < N) atomicAdd(&counts[batch[n]], 1.f);
}

// ---------------- out = tanh((gsums/max(counts,1)) @ Wfct + bfc), 512x64, K=128
// block: 128 threads (4 waves), 16 graphs per block
__global__ void graph_fc(const float* __restrict__ gsums, const float* __restrict__ counts,
                         const float* __restrict__ Wfct, const float* __restrict__ bfc,
                         float* __restrict__ out, int G) {
  __shared__ float As[16 * 132];
  const int tid = threadIdx.x;
  const int g0 = blockIdx.x * 16;
  for (int i = tid; i < 16 * HID; i += 128) {
    int r = i >> 7, k = i & 127;
    int g = g0 + r;
    float v = 0.f;
    if (g < G) v = gsums[(size_t)g * HID + k] / fmaxf(counts[g], 1.f);
    As[r * 132 + k] = v;
  }
  __syncthreads();
  const int lane = tid & 31, wave = tid >> 5;  // wave 0..3 -> cols [16w,16w+16)
  const int half = lane >> 4, lm = lane & 15;
  const int ncol = wave * 16 + lm;
  v8f c = {};
#pragma unroll
  for (int kk = 0; kk < HID; kk += 4) {
    int kr = kk + 2 * half;
    v2f a, b;
    a.x = As[lm * 132 + kr];
    a.y = As[lm * 132 + kr + 1];
    b.x = Wfct[kr * EMB_D + ncol];
    b.y = Wfct[(kr + 1) * EMB_D + ncol];
    c = wmma_f32_k4(a, b, c);
  }
#pragma unroll
  for (int r = 0; r < 8; ++r) {
    int g = g0 + r + 8 * half;
    if (g < G) out[(size_t)g * EMB_D + ncol] = tanhf(c[r] + bfc[ncol]);
  }
}

extern "C" void kernel_launch(void* const* d_in, const int* in_sizes, int n_in,
                              void* d_out, int out_size, void* d_ws, size_t ws_size,
                              hipStream_t stream) {
  const float* x     = (const float*)d_in[0];
  const int*   ei    = (const int*)d_in[1];   // [2][E]: src row then dst row
  const int*   rev   = (const int*)d_in[2];
  const float* eattr = (const float*)d_in[3];
  const int*   batch = (const int*)d_in[4];
  const float* W1    = (const float*)d_in[6];
  const float* W2    = (const float*)d_in[7];
  const float* W3    = (const float*)d_in[8];
  const float* b3    = (const float*)d_in[9];
  const float* Wfc   = (const float*)d_in[10];
  const float* bfc   = (const float*)d_in[11];
  float* out = (float*)d_out;

  const int E = in_sizes[3];          // edge_attr count
  const int N = in_sizes[4];          // batch count (= num_nodes)
  const int G = out_size / EMB_D;
  const int* src = ei;
  const int* dst = ei + E;

  // workspace layout (256B aligned slots)
  char* ws = (char*)d_ws;
  size_t off = 0;
  auto alloc = [&](size_t bytes) {
    float* p = (float*)(ws + off);
    off += (bytes + 255) & ~(size_t)255;
    return p;
  };
  float* h0     = alloc((size_t)E * HID * 4);   // persistent h0 (and final h)
  float* hA     = alloc((size_t)E * HID * 4);   // ping buffer
  float* m_node = alloc((size_t)N * HID * 4);   // per-node aggregate
  float* gsums  = alloc((size_t)G * HID * 4);
  float* counts = alloc((size_t)G * 4);
  float* W1t    = alloc((size_t)136 * HID * 4);
  float* W2t    = alloc((size_t)HID * HID * 4);
  float* W3t    = alloc((size_t)264 * HID * 4);
  float* Wfct   = alloc((size_t)HID * EMB_D * 4);
  (void)ws_size; (void)n_in;

  prep_weights<<<(264 * HID + 255) / 256, 256, 0, stream>>>(W1, W2, W3, Wfc, W1t, W2t, W3t, Wfct);

  const int eblocks = (E + 63) / 64;
  const int mn = N * HID;
  const int segthreads = E * 32;

  edge_init_gemm<<<eblocks, 256, 0, stream>>>(x, src, eattr, W1t, h0, E);

  // message passing iteration 1: h0 -> hA
  zero_f32<<<(mn + 255) / 256, 256, 0, stream>>>(m_node, mn);
  edge_segsum<<<(segthreads + 255) / 256, 256, 0, stream>>>(h0, dst, m_node, E);
  edge_msg_gemm<<<eblocks, 256, 0, stream>>>(h0, m_node, src, rev, W2t, h0, hA, E);

  // message passing iteration 2: hA -> h0 (in-place vs. h0 additive is block-local, safe)
  zero_f32<<<(mn + 255) / 256, 256, 0, stream>>>(m_node, mn);
  edge_segsum<<<(segthreads + 255) / 256, 256, 0, stream>>>(hA, dst, m_node, E);
  edge_msg_gemm<<<eblocks, 256, 0, stream>>>(hA, m_node, src, rev, W2t, h0, h0, E);

  // final per-node aggregation: v_msg = segsum(h_final)
  zero_f32<<<(mn + 255) / 256, 256, 0, stream>>>(m_node, mn);
  edge_segsum<<<(segthreads + 255) / 256, 256, 0, stream>>>(h0, dst, m_node, E);

  // node transform + mean pool
  zero_f32<<<(G * HID + 255) / 256, 256, 0, stream>>>(gsums, G * HID);
  zero_f32<<<(G + 255) / 256, 256, 0, stream>>>(counts, G);
  node_counts<<<(N + 255) / 256, 256, 0, stream>>>(batch, counts, N);
  node_gemm_pool<<<(N + 63) / 64, 256, 0, stream>>>(x, m_node, W3t, b3, batch, gsums, N);

  // readout
  graph_fc<<<(G + 15) / 16, 128, 0, stream>>>(gsums, counts, Wfct, bfc, out, G);
}